// SizePenaltyLoss_28518582846343
// MI455X (gfx1250) — compile-verified
//
#include <hip/hip_runtime.h>
#include <stdint.h>

#ifndef __has_builtin
#define __has_builtin(x) 0
#endif

#if defined(__AMDGCN__) && __has_builtin(__builtin_amdgcn_wmma_f32_16x16x4_f32)
#define USE_WMMA_REDUCE 1
#else
#define USE_WMMA_REDUCE 0
#endif

typedef __attribute__((ext_vector_type(2))) float v2f;
typedef __attribute__((ext_vector_type(8))) float v8f;

#define IMG_W   1024u
#define IMG_H   1024u
#define IMG_HW  (IMG_W * IMG_H)          // 1,048,576
#define NBATCH  8u
#define NPIX    (NBATCH * IMG_HW)        // 8,388,608
#define NOLBL   0xFFFFFFFFu

// ---------------------------------------------------------------------------
// Lock-free union-find (min-root), agent-scope relaxed loads so concurrent
// atomicMin updates from other WGPs are observed.
// ---------------------------------------------------------------------------
__device__ __forceinline__ unsigned ld_rlx(const unsigned* p) {
  return __hip_atomic_load(p, __ATOMIC_RELAXED, __HIP_MEMORY_SCOPE_AGENT);
}

__device__ __forceinline__ unsigned ccl_find(unsigned* L, unsigned x) {
  unsigned n = ld_rlx(&L[x]);
  while (n != x) { x = n; n = ld_rlx(&L[x]); }
  return x;
}

__device__ __forceinline__ unsigned ccl_find_ro(const unsigned* __restrict__ L, unsigned x) {
  unsigned n = L[x];
  while (n != x) { x = n; n = L[x]; }
  return x;
}

__device__ __forceinline__ void ccl_union(unsigned* L, unsigned a, unsigned b) {
  while (true) {
    a = ccl_find(L, a);
    b = ccl_find(L, b);
    if (a == b) return;
    if (a < b) { unsigned t = a; a = b; b = t; }   // a = larger index
    unsigned old = atomicMin(&L[a], b);
    if (old == a) return;                          // linked root a -> b
    a = old;                                       // lost race; merge old chain
  }
}

// ---------------------------------------------------------------------------
// K0: zero the two scalar accumulators (ce_sum, pen_sum)
// ---------------------------------------------------------------------------
__global__ void k_zero_accum(float* __restrict__ accum) {
  if (threadIdx.x < 2) accum[threadIdx.x] = 0.0f;
}

// ---------------------------------------------------------------------------
// K1: fused streaming pass — per-pixel CE, argmax mask, label init, bin zero.
// CE values are accumulated on the matrix pipe: ones(16x4) x B(4x16) column
// sums via V_WMMA_F32_16X16X4_F32 (64 f32 adds per instruction, f32 accum).
// ---------------------------------------------------------------------------
__device__ __forceinline__ float pixel_ce_init(const float* __restrict__ pred,
                                               const int* __restrict__ tgt,
                                               unsigned* __restrict__ labels,
                                               unsigned* __restrict__ sizes,
                                               unsigned p) {
  unsigned b = p >> 20;                 // p / IMG_HW
  unsigned r = p & (IMG_HW - 1u);       // p % IMG_HW
  const float* base = pred + (size_t)b * 2u * IMG_HW;
  float x0 = base[r];
  float x1 = base[IMG_HW + r];
  labels[p] = (x1 > x0) ? r : NOLBL;    // argmax==1 (ties -> class 0, like jnp.argmax)
  sizes[p]  = 0u;                       // zero the size bins in the same sweep
  float mx  = fmaxf(x0, x1);
  float lse = mx + logf(expf(x0 - mx) + expf(x1 - mx));
  float xt  = tgt[p] ? x1 : x0;
  return lse - xt;                      // -log_softmax[target]
}

__global__ void k_ce_init(const float* __restrict__ pred, const int* __restrict__ tgt,
                          unsigned* __restrict__ labels, unsigned* __restrict__ sizes,
                          float* __restrict__ accum) {
  const unsigned S   = gridDim.x * blockDim.x;              // 524288
  const unsigned tid = blockIdx.x * blockDim.x + threadIdx.x;
#if USE_WMMA_REDUCE
  v8f acc = {};
  v2f ones; ones[0] = 1.0f; ones[1] = 1.0f;                 // A = ones(16x4)
#else
  float accs = 0.0f;
#endif
  // NPIX is an exact multiple of 2*S: every lane runs every iteration,
  // EXEC stays all-ones through each WMMA (ISA requirement).
  for (unsigned p0 = tid; p0 < NPIX; p0 += 2u * S) {
    unsigned p1 = p0 + S;
    float c0 = pixel_ce_init(pred, tgt, labels, sizes, p0);
    float c1 = pixel_ce_init(pred, tgt, labels, sizes, p1);
#if USE_WMMA_REDUCE
    v2f bv; bv[0] = c0; bv[1] = c1;                         // B = 4x16 of CE values
    acc = __builtin_amdgcn_wmma_f32_16x16x4_f32(
        /*neg_a=*/false, ones, /*neg_b=*/false, bv,
        /*c_mod=*/(short)0, acc, /*reuse_a=*/false, /*reuse_b=*/false);
#else
    accs += c0 + c1;
#endif
  }
#if USE_WMMA_REDUCE
  // Every row of D equals the column sums; c[0] holds rows M=0 (lanes 0-15)
  // and M=8 (lanes 16-31), i.e. the total appears twice across the wave.
  float v = acc[0] * 0.5f;
#else
  float v = accs;
#endif
  for (int m = 16; m >= 1; m >>= 1) v += __shfl_xor(v, m, 32);
  if ((threadIdx.x & 31u) == 0u) atomicAdd(&accum[0], v);
}

// ---------------------------------------------------------------------------
// K2: single merge pass over right/down edges (4-connectivity).
// ---------------------------------------------------------------------------
__global__ void k_merge(unsigned* __restrict__ labels) {
  unsigned p = blockIdx.x * blockDim.x + threadIdx.x;
  if (p >= NPIX) return;
  if (labels[p] == NOLBL) return;           // background never changes
  unsigned b = p >> 20;
  unsigned r = p & (IMG_HW - 1u);
  unsigned y = r >> 10, x = r & (IMG_W - 1u);
  unsigned* L = labels + ((size_t)b << 20);
  // Foreground labels are always < IMG_HW, background stays NOLBL forever,
  // so a plain load is a stable foreground test even mid-merge.
  if (x + 1u < IMG_W && L[r + 1u]    != NOLBL) ccl_union(L, r, r + 1u);
  if (y + 1u < IMG_H && L[r + IMG_W] != NOLBL) ccl_union(L, r, r + IMG_W);
}

// ---------------------------------------------------------------------------
// K3: scatter-add component sizes at the (min-index) roots.
// ---------------------------------------------------------------------------
__global__ void k_count(const unsigned* __restrict__ labels, unsigned* __restrict__ sizes) {
  unsigned p = blockIdx.x * blockDim.x + threadIdx.x;
  if (p >= NPIX) return;
  if (labels[p] == NOLBL) return;
  unsigned b = p >> 20;
  unsigned r = p & (IMG_HW - 1u);
  const unsigned* L = labels + ((size_t)b << 20);
  unsigned root = ccl_find_ro(L, r);
  atomicAdd(&sizes[((size_t)b << 20) + root], 1u);
}

// ---------------------------------------------------------------------------
// K4: size penalty, grid-stride + wave/block reduce, one atomic per block.
// ---------------------------------------------------------------------------
__global__ void k_penalty(const unsigned* __restrict__ sizes, float* __restrict__ accum) {
  const unsigned S = gridDim.x * blockDim.x;
  float local = 0.0f;
  for (unsigned i = blockIdx.x * blockDim.x + threadIdx.x; i < NPIX; i += S) {
    unsigned s = sizes[i];
    if (s != 0u) {
      float fs = (float)s;
      if (fs < 50.0f)       local += (50.0f - fs) * (1.0f / 50.0f);
      else if (fs > 500.0f) local += (fs - 500.0f) * (1.0f / 500.0f);
    }
  }
  for (int m = 16; m >= 1; m >>= 1) local += __shfl_xor(local, m, 32);
  __shared__ float wsum[8];
  unsigned wid = threadIdx.x >> 5, lane = threadIdx.x & 31u;
  if (lane == 0u) wsum[wid] = local;
  __syncthreads();
  if (threadIdx.x == 0u) {
    float t = 0.0f;
    for (unsigned i = 0; i < (blockDim.x >> 5); ++i) t += wsum[i];
    atomicAdd(&accum[1], t);
  }
}

// ---------------------------------------------------------------------------
// K5: finalize: ce_mean + 0.1 * (pen_sum / B)
// ---------------------------------------------------------------------------
__global__ void k_final(const float* __restrict__ accum, float* __restrict__ out) {
  if (blockIdx.x == 0 && threadIdx.x == 0) {
    out[0] = accum[0] * (1.0f / (float)NPIX) + 0.1f * (accum[1] * (1.0f / (float)NBATCH));
  }
}

// ---------------------------------------------------------------------------
extern "C" void kernel_launch(void* const* d_in, const int* in_sizes, int n_in,
                              void* d_out, int out_size, void* d_ws, size_t ws_size,
                              hipStream_t stream) {
  (void)in_sizes; (void)n_in; (void)out_size; (void)ws_size;
  const float* pred = (const float*)d_in[0];   // [8, 2, 1024, 1024] f32
  const int*   tgt  = (const int*)d_in[1];     // [8, 1024, 1024] i32
  float*       out  = (float*)d_out;           // scalar f32

  // Workspace layout: labels (32MB) | sizes (32MB) | accum (2 floats)
  unsigned* labels = (unsigned*)d_ws;
  unsigned* sizes  = labels + NPIX;
  float*    accum  = (float*)(sizes + NPIX);

  k_zero_accum<<<1, 64, 0, stream>>>(accum);
  k_ce_init  <<<2048, 256, 0, stream>>>(pred, tgt, labels, sizes, accum);
  k_merge    <<<NPIX / 256, 256, 0, stream>>>(labels);
  k_count    <<<NPIX / 256, 256, 0, stream>>>(labels, sizes);
  k_penalty  <<<2048, 256, 0, stream>>>(sizes, accum);
  k_final    <<<1, 64, 0, stream>>>(accum, out);
}